// MSN_5059471475050
// MI455X (gfx1250) — compile-verified
//
#include <hip/hip_runtime.h>

typedef float v2f __attribute__((ext_vector_type(2)));
typedef float v8f __attribute__((ext_vector_type(8)));

constexpr int BB = 32, NN = 8192, CC = 64, NPP = 32;
constexpr float EPS = 1e-5f;

// workspace layout (float offsets)
constexpr size_t WS_W1A  = 0;        // 4 tiles * 32 lanes * 2   = 256 floats
constexpr size_t WS_BF1  = 256;      // 64
constexpr size_t WS_W2A  = 320;      // 8*16 tiles * 64          = 8192
constexpr size_t WS_BF2  = 8512;     // 128
constexpr size_t WS_W3A  = 8640;     // 4*32 tiles * 64          = 8192
constexpr size_t WS_BF3  = 16832;    // 64
constexpr size_t WS_SIDX = 16896;    // B*C*NP ints              = 65536
constexpr size_t WS_FEAT = 82432;    // B*C*N floats             = 16777216

// ---------------------------------------------------------------------------
// Fold BN into conv weights/bias and pre-swizzle A-operands into the
// CDNA5 fp32 16x4 A-matrix layout: lane<16 holds K=j (j=0,1), lane>=16 holds K=j+2.
// ---------------------------------------------------------------------------
__global__ void fold_kernel(const float* __restrict__ W1, const float* __restrict__ b1,
                            const float* __restrict__ g1, const float* __restrict__ be1,
                            const float* __restrict__ m1, const float* __restrict__ v1,
                            const float* __restrict__ W2, const float* __restrict__ b2,
                            const float* __restrict__ g2, const float* __restrict__ be2,
                            const float* __restrict__ m2, const float* __restrict__ v2,
                            const float* __restrict__ W3, const float* __restrict__ b3,
                            const float* __restrict__ g3, const float* __restrict__ be3,
                            const float* __restrict__ m3, const float* __restrict__ v3,
                            float* __restrict__ ws) {
    const int tid = threadIdx.x;
    for (int o = tid; o < 64; o += blockDim.x) {
        float a1 = g1[o] * rsqrtf(v1[o] + EPS);
        ws[WS_BF1 + o] = a1 * b1[o] + be1[o] - m1[o] * a1;
        float a3 = g3[o] * rsqrtf(v3[o] + EPS);
        ws[WS_BF3 + o] = a3 * b3[o] + be3[o] - m3[o] * a3;
    }
    for (int o = tid; o < 128; o += blockDim.x) {
        float a2 = g2[o] * rsqrtf(v2[o] + EPS);
        ws[WS_BF2 + o] = a2 * b2[o] + be2[o] - m2[o] * a2;
    }
    // Layer 1: 4 out-tiles, 1 k-step (3 in-ch padded to 4)
    for (int i = tid; i < 4 * 32 * 2; i += blockDim.x) {
        int j = i & 1, l = (i >> 1) & 31, tm = i >> 6;
        int mm = tm * 16 + (l & 15);
        int ks = (l < 16) ? j : j + 2;
        float a = g1[mm] * rsqrtf(v1[mm] + EPS);
        ws[WS_W1A + i] = (ks < 3) ? a * W1[mm * 3 + ks] : 0.0f;
    }
    // Layer 2: tiles indexed tk = tm*16 + kk  (tm 0..7, kk 0..15)
    for (int i = tid; i < 8192; i += blockDim.x) {
        int j = i & 1, l = (i >> 1) & 31, tk = i >> 6;
        int kk = tk & 15, tm = tk >> 4;
        int mm = tm * 16 + (l & 15);
        int kch = kk * 4 + ((l < 16) ? j : j + 2);
        float a = g2[mm] * rsqrtf(v2[mm] + EPS);
        ws[WS_W2A + i] = a * W2[mm * 64 + kch];
    }
    // Layer 3: tiles indexed tk = tm*32 + kk  (tm 0..3, kk 0..31)
    for (int i = tid; i < 8192; i += blockDim.x) {
        int j = i & 1, l = (i >> 1) & 31, tk = i >> 6;
        int kk = tk & 31, tm = tk >> 5;
        int mm = tm * 16 + (l & 15);
        int kch = kk * 4 + ((l < 16) ? j : j + 2);
        float a = g3[mm] * rsqrtf(v3[mm] + EPS);
        ws[WS_W3A + i] = a * W3[mm * 128 + kch];
    }
}

// ---------------------------------------------------------------------------
// MLP via V_WMMA_F32_16X16X4_F32. One wave = one 16-point tile through all
// three layers. Activations staged per-wave in LDS (ch-major, conflict-free),
// re-read as B operands. B layout assumed mirror of A: VGPR0 = rows K=0|K=2,
// VGPR1 = rows K=1|K=3 across lane halves (matches C/D striping pattern).
// ---------------------------------------------------------------------------
__global__ __launch_bounds__(128) void mlp_kernel(const float* __restrict__ partial,
                                                  const float* __restrict__ ws,
                                                  float* __restrict__ feat) {
    __shared__ float h1s[4][64 * 16];   // 16 KB
    __shared__ float h2s[4][128 * 16];  // 32 KB
    const int lane = threadIdx.x & 31;
    const int wave = threadIdx.x >> 5;
    const int pt   = lane & 15;
    const int half = lane >> 4;
    const int b    = blockIdx.x >> 7;                       // 128 blocks per batch
    const int n0   = ((blockIdx.x & 127) << 6) + wave * 16; // 64 points per block

    const v2f* W1A = (const v2f*)(ws + WS_W1A);
    const v2f* W2A = (const v2f*)(ws + WS_W2A);
    const v2f* W3A = (const v2f*)(ws + WS_W3A);
    const float* BF1 = ws + WS_BF1;
    const float* BF2 = ws + WS_BF2;
    const float* BF3 = ws + WS_BF3;

    // ---- Layer 1: B operand straight from `partial` (3 ch padded to 4) ----
    const float* pin = partial + (size_t)b * 3 * NN + n0 + pt;
    v2f Bv1;
    Bv1.x = half ? pin[2 * (size_t)NN] : pin[0];
    Bv1.y = half ? 0.0f : pin[(size_t)NN];

    const v8f vzero = {0.f, 0.f, 0.f, 0.f, 0.f, 0.f, 0.f, 0.f};
#pragma unroll
    for (int tm = 0; tm < 4; ++tm) {
        v2f Av = W1A[tm * 32 + lane];
        v8f acc = __builtin_amdgcn_wmma_f32_16x16x4_f32(
            false, Av, false, Bv1, (short)0, vzero, false, false);
#pragma unroll
        for (int r = 0; r < 8; ++r) {
            int ch = tm * 16 + half * 8 + r;
            h1s[wave][ch * 16 + pt] = fmaxf(acc[r] + BF1[ch], 0.0f);
        }
    }
    __syncthreads();

    // ---- Layer 2: 64 -> 128, 16 k-steps x 8 out-tiles = 128 WMMA ----
    v8f acc2[8];
#pragma unroll
    for (int t = 0; t < 8; ++t) acc2[t] = vzero;
#pragma unroll 4
    for (int kk = 0; kk < 16; ++kk) {
        v2f Bv;
        Bv.x = h1s[wave][(kk * 4 + half * 2) * 16 + pt];
        Bv.y = h1s[wave][(kk * 4 + 1 + half * 2) * 16 + pt];
#pragma unroll
        for (int tm = 0; tm < 8; ++tm) {
            v2f Av = W2A[(tm * 16 + kk) * 32 + lane];
            acc2[tm] = __builtin_amdgcn_wmma_f32_16x16x4_f32(
                false, Av, false, Bv, (short)0, acc2[tm], false, false);
        }
    }
#pragma unroll
    for (int tm = 0; tm < 8; ++tm) {
#pragma unroll
        for (int r = 0; r < 8; ++r) {
            int ch = tm * 16 + half * 8 + r;
            h2s[wave][ch * 16 + pt] = fmaxf(acc2[tm][r] + BF2[ch], 0.0f);
        }
    }
    __syncthreads();

    // ---- Layer 3: 128 -> 64, 32 k-steps x 4 out-tiles = 128 WMMA ----
    v8f acc3[4];
#pragma unroll
    for (int t = 0; t < 4; ++t) acc3[t] = vzero;
#pragma unroll 4
    for (int kk = 0; kk < 32; ++kk) {
        v2f Bv;
        Bv.x = h2s[wave][(kk * 4 + half * 2) * 16 + pt];
        Bv.y = h2s[wave][(kk * 4 + 1 + half * 2) * 16 + pt];
#pragma unroll
        for (int tm = 0; tm < 4; ++tm) {
            v2f Av = W3A[(tm * 32 + kk) * 32 + lane];
            acc3[tm] = __builtin_amdgcn_wmma_f32_16x16x4_f32(
                false, Av, false, Bv, (short)0, acc3[tm], false, false);
        }
    }
    float* fout = feat + (size_t)b * 64 * NN + n0 + pt;
#pragma unroll
    for (int tm = 0; tm < 4; ++tm) {
#pragma unroll
        for (int r = 0; r < 8; ++r) {
            int ch = tm * 16 + half * 8 + r;
            fout[(size_t)ch * NN] = acc3[tm][r] + BF3[ch];  // no ReLU on layer 3
        }
    }
}

// ---------------------------------------------------------------------------
// Stable argsort per (b,c) row: bitonic sort of packed (sortable-key | index)
// u64 in LDS (64 KB / row), then emit the 32 sampled order statistics.
// ---------------------------------------------------------------------------
__global__ __launch_bounds__(1024) void sort_kernel(const float* __restrict__ feat,
                                                    int* __restrict__ sidx) {
    __shared__ unsigned long long s[NN];  // 64 KB
    const int row = blockIdx.x;           // b*64 + c
    const int tid = threadIdx.x;
    const float* f = feat + (size_t)row * NN;
    for (int t = tid; t < NN; t += 1024) {
        unsigned u = __float_as_uint(f[t]);
        u = (u & 0x80000000u) ? ~u : (u | 0x80000000u);  // monotonic float->uint
        s[t] = ((unsigned long long)u << 32) | (unsigned)t;
    }
    __syncthreads();
    for (unsigned k = 2; k <= (unsigned)NN; k <<= 1) {
        for (unsigned j = k >> 1; j > 0; j >>= 1) {
            for (unsigned t = tid; t < (unsigned)NN; t += 1024) {
                unsigned p = t ^ j;
                if (p > t) {
                    unsigned long long a = s[t], b = s[p];
                    bool up = ((t & k) == 0);
                    if ((a > b) == up) { s[t] = b; s[p] = a; }
                }
            }
            __syncthreads();
        }
    }
    if (tid < NPP) {
        int pos = (tid * (NN - 1)) / (NPP - 1);  // floor(linspace(0, N-1, NP))
        sidx[row * NPP + tid] = (int)(unsigned)(s[pos] & 0xFFFFFFFFu);
    }
}

// ---------------------------------------------------------------------------
// Gather: out[b,ch,c2,k] (ch<64: feat, ch>=64: partial) then sp_idx.
// ---------------------------------------------------------------------------
__global__ __launch_bounds__(256) void gather_kernel(const float* __restrict__ feat,
                                                     const float* __restrict__ partial,
                                                     const int* __restrict__ sidx,
                                                     float* __restrict__ out) {
    const int OUT1 = BB * 67 * CC * NPP;           // 4,390,912
    const int TOTAL = OUT1 + BB * 3 * CC * NPP;    // 4,587,520
    int gid = blockIdx.x * blockDim.x + threadIdx.x;
    if (gid >= TOTAL) return;
    if (gid < OUT1) {
        int k = gid & 31, c2 = (gid >> 5) & 63;
        int rest = gid >> 11;
        int ch = rest % 67, b = rest / 67;
        int n = sidx[(b * CC + c2) * NPP + k];
        float v = (ch < 64)
            ? feat[((size_t)(b * CC + ch)) * NN + n]
            : partial[((size_t)(b * 3 + (ch - 64))) * NN + n];
        out[gid] = v;
    } else {
        int g2 = gid - OUT1;
        int k = g2 & 31, c2 = (g2 >> 5) & 63;
        int rest = g2 >> 11;
        int b = rest / 3;  // j index unused: sp_idx broadcast over j
        out[gid] = (float)sidx[(b * CC + c2) * NPP + k];
    }
}

extern "C" void kernel_launch(void* const* d_in, const int* in_sizes, int n_in,
                              void* d_out, int out_size, void* d_ws, size_t ws_size,
                              hipStream_t stream) {
    (void)in_sizes; (void)n_in; (void)out_size; (void)ws_size;
    const float* partial = (const float*)d_in[0];
    const float* W1 = (const float*)d_in[1];  const float* b1 = (const float*)d_in[2];
    const float* g1 = (const float*)d_in[3];  const float* be1 = (const float*)d_in[4];
    const float* m1 = (const float*)d_in[5];  const float* v1 = (const float*)d_in[6];
    const float* W2 = (const float*)d_in[7];  const float* b2 = (const float*)d_in[8];
    const float* g2 = (const float*)d_in[9];  const float* be2 = (const float*)d_in[10];
    const float* m2 = (const float*)d_in[11]; const float* v2 = (const float*)d_in[12];
    const float* W3 = (const float*)d_in[13]; const float* b3 = (const float*)d_in[14];
    const float* g3 = (const float*)d_in[15]; const float* be3 = (const float*)d_in[16];
    const float* m3 = (const float*)d_in[17]; const float* v3 = (const float*)d_in[18];

    float* ws = (float*)d_ws;
    float* feat = ws + WS_FEAT;
    int* sidx = (int*)(ws + WS_SIDX);

    fold_kernel<<<1, 256, 0, stream>>>(W1, b1, g1, be1, m1, v1,
                                       W2, b2, g2, be2, m2, v2,
                                       W3, b3, g3, be3, m3, v3, ws);
    mlp_kernel<<<BB * NN / 64, 128, 0, stream>>>(partial, ws, feat);
    sort_kernel<<<BB * CC, 1024, 0, stream>>>(feat, sidx);
    const int TOTAL = BB * 67 * CC * NPP + BB * 3 * CC * NPP;
    gather_kernel<<<(TOTAL + 255) / 256, 256, 0, stream>>>(feat, partial, sidx, (float*)d_out);
}